// GNNEncoderWithEdges_6914897347058
// MI455X (gfx1250) — compile-verified
//
#include <hip/hip_runtime.h>
#include <hip/hip_bf16.h>

// ---------------------------------------------------------------------------
// GINE-style GNN encoder for MI455X (gfx1250, wave32).
// Dense GEMMs: V_WMMA_F32_16X16X4_F32 (exact fp32), multi-row-tile per wave
// so each B (weight) fragment feeds 2-4 WMMAs -> 2-4x less L2 weight traffic.
// Memory-bound phases (edge stream / gather / scatter-add) use coalesced
// float4 traffic; h/z (51 MB) stay L2-resident on the 192 MB L2.
// ---------------------------------------------------------------------------

typedef __attribute__((ext_vector_type(2))) float v2f;
typedef __attribute__((ext_vector_type(8))) float v8f;

#define HD  128   // hidden dim
#define FND 64    // node feature dim
#define FED 16    // edge feature dim
#define PAD 4     // LDS row padding (floats) -> conflict-free strided reads

static __device__ __forceinline__ v8f wmma_f32_4(v2f a, v2f b, v8f c) {
    // D = A(16x4, f32) * B(4x16, f32) + C(16x16, f32)
    return __builtin_amdgcn_wmma_f32_16x16x4_f32(
        /*neg_a=*/false, a, /*neg_b=*/false, b,
        /*c_mod=*/(short)0, c, /*reuse_a=*/false, /*reuse_b=*/false);
}

// ---------------------------------------------------------------------------
// Kernel 1: h = relu(X[N,64] @ Wn[64,128] + bn)
// 32 rows/block (grid.x = N/32), 8 waves; each wave: 2 row-tiles x 1 col-tile.
// ---------------------------------------------------------------------------
__global__ __launch_bounds__(256, 2)
void node_proj_kernel(const float* __restrict__ X, const float* __restrict__ Wn,
                      const float* __restrict__ bn, float* __restrict__ Hout) {
    __shared__ __align__(16) float As[32][FND + PAD];   // 32 x 64 stripe

    const int wave = threadIdx.x >> 5;
    const int lane = threadIdx.x & 31;
    const int half = lane >> 4;
    const int l16  = lane & 15;
    const int row0 = blockIdx.x * 32;

    // stage 32x64 A stripe: 2048 floats, 256 threads x 8 floats (2x float4)
    {
        int idx = threadIdx.x * 8;
        int r = idx >> 6;                   // /64
        int c = idx & 63;
        const float4* p = (const float4*)(X + (size_t)(row0 + r) * FND + c);
        *(float4*)(&As[r][c + 0]) = p[0];
        *(float4*)(&As[r][c + 4]) = p[1];
    }
    __syncthreads();

    const int colbase = wave * 16;
    v8f acc0 = {}, acc1 = {};
#pragma unroll
    for (int k = 0; k < FND; k += 4) {
        v2f b;
        b.x = Wn[(size_t)(k + half * 2 + 0) * HD + colbase + l16];
        b.y = Wn[(size_t)(k + half * 2 + 1) * HD + colbase + l16];
        v2f a0 = *(const v2f*)(&As[l16][k + half * 2]);
        v2f a1 = *(const v2f*)(&As[16 + l16][k + half * 2]);
        acc0 = wmma_f32_4(a0, b, acc0);
        acc1 = wmma_f32_4(a1, b, acc1);
    }

    const int col  = colbase + l16;
    const float bias = bn[col];
#pragma unroll
    for (int v = 0; v < 8; ++v) {
        int r = v + half * 8;
        Hout[(size_t)(row0 + r) * HD + col]      = fmaxf(acc0[v] + bias, 0.0f);
        Hout[(size_t)(row0 + 16 + r) * HD + col] = fmaxf(acc1[v] + bias, 0.0f);
    }
}

// ---------------------------------------------------------------------------
// Kernel 2: e = relu(Xe[E,16] @ We1 + be1) @ We2 + be2
// 64 edges/block (grid.x = E/64); each wave: 4 row-tiles x 1 col-tile,
// so every We2 fragment load feeds 4 WMMAs (4x less L2 weight traffic).
// ---------------------------------------------------------------------------
__global__ __launch_bounds__(256, 2)
void edge_mlp_kernel(const float* __restrict__ Xe,
                     const float* __restrict__ We1, const float* __restrict__ be1,
                     const float* __restrict__ We2, const float* __restrict__ be2,
                     float* __restrict__ Eout) {
    __shared__ __align__(16) float Xs[64][FED + PAD];  // 64 x 16 edge feats
    __shared__ __align__(16) float Ts[64][HD + PAD];   // 64 x 128 relu intermediate

    const int wave = threadIdx.x >> 5;
    const int lane = threadIdx.x & 31;
    const int half = lane >> 4;
    const int l16  = lane & 15;
    const int e0   = blockIdx.x * 64;

    // stage 64x16 edge-feature tile: 1024 floats, 256 threads x float4
    {
        int idx = threadIdx.x * 4;
        int r = idx >> 4;
        int c = idx & 15;
        *(float4*)(&Xs[r][c]) = *(const float4*)(Xe + (size_t)(e0 + r) * FED + c);
    }
    __syncthreads();

    const int colbase = wave * 16;

    // stage 1: t = relu(Xe @ We1 + be1), K = 16
    {
        v8f acc[4] = {{}, {}, {}, {}};
#pragma unroll
        for (int k = 0; k < FED; k += 4) {
            v2f b;
            b.x = We1[(size_t)(k + half * 2 + 0) * HD + colbase + l16];
            b.y = We1[(size_t)(k + half * 2 + 1) * HD + colbase + l16];
#pragma unroll
            for (int rt = 0; rt < 4; ++rt) {
                v2f a = *(const v2f*)(&Xs[rt * 16 + l16][k + half * 2]);
                acc[rt] = wmma_f32_4(a, b, acc[rt]);
            }
        }
        float bias = be1[colbase + l16];
#pragma unroll
        for (int rt = 0; rt < 4; ++rt) {
#pragma unroll
            for (int v = 0; v < 8; ++v) {
                int r = rt * 16 + v + half * 8;
                Ts[r][colbase + l16] = fmaxf(acc[rt][v] + bias, 0.0f);
            }
        }
    }
    __syncthreads();

    // stage 2: e = t @ We2 + be2, K = 128
    {
        v8f acc[4] = {{}, {}, {}, {}};
#pragma unroll 4
        for (int k = 0; k < HD; k += 4) {
            v2f b;
            b.x = We2[(size_t)(k + half * 2 + 0) * HD + colbase + l16];
            b.y = We2[(size_t)(k + half * 2 + 1) * HD + colbase + l16];
#pragma unroll
            for (int rt = 0; rt < 4; ++rt) {
                v2f a = *(const v2f*)(&Ts[rt * 16 + l16][k + half * 2]);
                acc[rt] = wmma_f32_4(a, b, acc[rt]);
            }
        }
        float bias = be2[colbase + l16];
#pragma unroll
        for (int rt = 0; rt < 4; ++rt) {
#pragma unroll
            for (int v = 0; v < 8; ++v) {
                int r = rt * 16 + v + half * 8;
                Eout[(size_t)(e0 + r) * HD + colbase + l16] = acc[rt][v] + bias;
            }
        }
    }
}

// ---------------------------------------------------------------------------
// Kernel 3: z <- h   (seed for (1+eps)*h + agg, eps = 0), float4 copy
// ---------------------------------------------------------------------------
__global__ __launch_bounds__(256)
void copy_kernel(const float* __restrict__ src, float* __restrict__ dst, long n4) {
    long i = (long)blockIdx.x * blockDim.x + threadIdx.x;
    if (i < n4) ((float4*)dst)[i] = ((const float4*)src)[i];
}

// ---------------------------------------------------------------------------
// Kernel 4: z[dst] += relu(h[src] + e)  — one wave per edge, lanes = cols/4.
// Edge id is wave-uniform -> scalarize src/dst via readfirstlane.
// Prefetch the streaming e rows ahead of use (global_prefetch_b8).
// ---------------------------------------------------------------------------
__global__ __launch_bounds__(256)
void scatter_kernel(const float* __restrict__ Hin, const float* __restrict__ Ein,
                    const int* __restrict__ srcIdx, const int* __restrict__ dstIdx,
                    float* __restrict__ Z, int E) {
    int tid  = blockIdx.x * 256 + threadIdx.x;
    int edge = tid >> 5;
    if (edge >= E) return;
    int lane = tid & 31;
    int col  = lane * 4;

    int s = __builtin_amdgcn_readfirstlane(srcIdx[edge]);
    int d = __builtin_amdgcn_readfirstlane(dstIdx[edge]);

    int pe = edge + 256;                       // stream-ahead hint for e
    if (pe < E) __builtin_prefetch(Ein + (size_t)pe * HD + col, 0, 1);

    float4 hv = *(const float4*)(Hin + (size_t)s * HD + col);
    float4 ev = *(const float4*)(Ein + (size_t)edge * HD + col);

    float m0 = fmaxf(hv.x + ev.x, 0.0f);
    float m1 = fmaxf(hv.y + ev.y, 0.0f);
    float m2 = fmaxf(hv.z + ev.z, 0.0f);
    float m3 = fmaxf(hv.w + ev.w, 0.0f);

    float* zp = Z + (size_t)d * HD + col;
    atomicAdd(zp + 0, m0);
    atomicAdd(zp + 1, m1);
    atomicAdd(zp + 2, m2);
    atomicAdd(zp + 3, m3);
}

// ---------------------------------------------------------------------------
// Kernel 5: h = relu( relu(z @ W1 + b1) @ W2 + b2 )
// 32 rows/block (grid.x = N/32); each wave: 2 row-tiles x 1 col-tile,
// halving W1/W2 L2 traffic vs 16-row blocks.
// ---------------------------------------------------------------------------
__global__ __launch_bounds__(256, 2)
void apply_mlp_kernel(const float* __restrict__ Z,
                      const float* __restrict__ W1, const float* __restrict__ b1,
                      const float* __restrict__ W2, const float* __restrict__ b2,
                      float* __restrict__ Hout) {
    __shared__ __align__(16) float Zs[32][HD + PAD];
    __shared__ __align__(16) float Ts[32][HD + PAD];

    const int wave = threadIdx.x >> 5;
    const int lane = threadIdx.x & 31;
    const int half = lane >> 4;
    const int l16  = lane & 15;
    const int row0 = blockIdx.x * 32;

    // stage 32x128 z tile: 4096 floats, 256 threads x 16 floats (4x float4)
    {
        int idx = threadIdx.x * 16;
        int r = idx >> 7;
        int c = idx & 127;
        const float4* p = (const float4*)(Z + (size_t)(row0 + r) * HD + c);
        *(float4*)(&Zs[r][c + 0])  = p[0];
        *(float4*)(&Zs[r][c + 4])  = p[1];
        *(float4*)(&Zs[r][c + 8])  = p[2];
        *(float4*)(&Zs[r][c + 12]) = p[3];
    }
    __syncthreads();

    const int colbase = wave * 16;

    // stage 1: t = relu(z @ W1 + b1), K = 128
    {
        v8f acc0 = {}, acc1 = {};
#pragma unroll 8
        for (int k = 0; k < HD; k += 4) {
            v2f b;
            b.x = W1[(size_t)(k + half * 2 + 0) * HD + colbase + l16];
            b.y = W1[(size_t)(k + half * 2 + 1) * HD + colbase + l16];
            v2f a0 = *(const v2f*)(&Zs[l16][k + half * 2]);
            v2f a1 = *(const v2f*)(&Zs[16 + l16][k + half * 2]);
            acc0 = wmma_f32_4(a0, b, acc0);
            acc1 = wmma_f32_4(a1, b, acc1);
        }
        float bias = b1[colbase + l16];
#pragma unroll
        for (int v = 0; v < 8; ++v) {
            int r = v + half * 8;
            Ts[r][colbase + l16]      = fmaxf(acc0[v] + bias, 0.0f);
            Ts[16 + r][colbase + l16] = fmaxf(acc1[v] + bias, 0.0f);
        }
    }
    __syncthreads();

    // stage 2: h = relu(t @ W2 + b2), K = 128
    {
        v8f acc0 = {}, acc1 = {};
#pragma unroll 8
        for (int k = 0; k < HD; k += 4) {
            v2f b;
            b.x = W2[(size_t)(k + half * 2 + 0) * HD + colbase + l16];
            b.y = W2[(size_t)(k + half * 2 + 1) * HD + colbase + l16];
            v2f a0 = *(const v2f*)(&Ts[l16][k + half * 2]);
            v2f a1 = *(const v2f*)(&Ts[16 + l16][k + half * 2]);
            acc0 = wmma_f32_4(a0, b, acc0);
            acc1 = wmma_f32_4(a1, b, acc1);
        }
        float bias = b2[colbase + l16];
#pragma unroll
        for (int v = 0; v < 8; ++v) {
            int r = v + half * 8;
            Hout[(size_t)(row0 + r) * HD + colbase + l16] =
                fmaxf(acc0[v] + bias, 0.0f);
            Hout[(size_t)(row0 + 16 + r) * HD + colbase + l16] =
                fmaxf(acc1[v] + bias, 0.0f);
        }
    }
}

// ---------------------------------------------------------------------------
// Host launcher
// inputs: node_feats, edge_feats, src, dst, Wn, bn, We1, be1, We2, be2,
//         W1[L,H,H], b1[L,H], W2[L,H,H], b2[L,H]
// ---------------------------------------------------------------------------
extern "C" void kernel_launch(void* const* d_in, const int* in_sizes, int n_in,
                              void* d_out, int out_size, void* d_ws, size_t ws_size,
                              hipStream_t stream) {
    const float* node_feats = (const float*)d_in[0];
    const float* edge_feats = (const float*)d_in[1];
    const int*   srcIdx     = (const int*)d_in[2];
    const int*   dstIdx     = (const int*)d_in[3];
    const float* Wn         = (const float*)d_in[4];
    const float* bn         = (const float*)d_in[5];
    const float* We1        = (const float*)d_in[6];
    const float* be1        = (const float*)d_in[7];
    const float* We2        = (const float*)d_in[8];
    const float* be2        = (const float*)d_in[9];
    const float* W1         = (const float*)d_in[10];
    const float* b1         = (const float*)d_in[11];
    const float* W2         = (const float*)d_in[12];
    const float* b2         = (const float*)d_in[13];

    const int N = in_sizes[0] / FND;   // 100000
    const int E = in_sizes[1] / FED;   // 600000
    const int L = in_sizes[11] / HD;   // 4

    float* hbuf  = (float*)d_out;             // persistent node state [N,H]
    float* e_buf = (float*)d_ws;              // [E,H]  edge embeddings
    float* z_buf = e_buf + (size_t)E * HD;    // [N,H]  h + aggregated messages

    // h0 = relu(X @ Wn + bn)
    node_proj_kernel<<<N / 32, 256, 0, stream>>>(node_feats, Wn, bn, hbuf);

    // e = relu(Xe @ We1 + be1) @ We2 + be2
    edge_mlp_kernel<<<E / 64, 256, 0, stream>>>(edge_feats, We1, be1, We2, be2, e_buf);

    const long n4 = (long)N * HD / 4;
    const int copy_blocks = (int)((n4 + 255) / 256);
    const int scat_blocks = (E * 32) / 256;

    for (int l = 0; l < L; ++l) {
        // z = h  (eps = 0 self term)
        copy_kernel<<<copy_blocks, 256, 0, stream>>>(hbuf, z_buf, n4);
        // z[dst] += relu(h[src] + e)
        scatter_kernel<<<scat_blocks, 256, 0, stream>>>(hbuf, e_buf, srcIdx, dstIdx,
                                                        z_buf, E);
        // h = relu(relu(z @ W1 + b1) @ W2 + b2)
        apply_mlp_kernel<<<N / 32, 256, 0, stream>>>(z_buf,
                                                     W1 + (size_t)l * HD * HD,
                                                     b1 + (size_t)l * HD,
                                                     W2 + (size_t)l * HD * HD,
                                                     b2 + (size_t)l * HD,
                                                     hbuf);
    }
}